// ProjectedEmbedding_33157147525314
// MI455X (gfx1250) — compile-verified
//
#include <hip/hip_runtime.h>

#define N_NODES 8192
#define TOPK    15

typedef float v2f __attribute__((ext_vector_type(2)));
typedef float v4f __attribute__((ext_vector_type(4)));
typedef float v8f __attribute__((ext_vector_type(8)));

// ---------------------------------------------------------------------------
// Kernel 1: per-node 2-layer MLP  ->  M2[n, d]   (memory-bound, 256 MB weights)
// 4 nodes per 256-thread block, 64 threads per node (one thread per output col)
// ---------------------------------------------------------------------------
__global__ __launch_bounds__(256) void mlp_kernel(
    const float* __restrict__ x,  const float* __restrict__ W1,
    const float* __restrict__ b1, const float* __restrict__ W2,
    const float* __restrict__ b2, float* __restrict__ M2) {
  __shared__ float xs[4][64];
  __shared__ float hs[4][64];
  const int local = threadIdx.x >> 6;
  const int t     = threadIdx.x & 63;
  const int node  = blockIdx.x * 4 + local;

  xs[local][t] = x[node * 64 + t];
  __syncthreads();

  const float* w1 = W1 + (size_t)node * 4096;
  float acc = b1[node * 64 + t];
#pragma unroll
  for (int f = 0; f < 64; ++f)
    acc = fmaf(xs[local][f], w1[f * 64 + t], acc);   // coalesced across t
  hs[local][t] = fmaxf(acc, 0.0f);
  __syncthreads();

  const float* w2 = W2 + (size_t)node * 4096;
  float acc2 = b2[node * 64 + t];
#pragma unroll
  for (int h = 0; h < 64; ++h)
    acc2 = fmaf(hs[local][h], w2[h * 64 + t], acc2);
  M2[node * 64 + t] = acc2;
}

// ---------------------------------------------------------------------------
// Kernel 2: A = relu(M1 @ M2^T - M2 @ M1^T) via V_WMMA_F32_16X16X4_F32.
// d[i][j] = dot(M1[i],M2[j]) - dot(M2[i],M1[j]) is antisymmetric, so only
// 32x32 blocks with bi <= bj are computed (2x2 tiles of 16x16 per wave);
// the mirror block (bj,bi) is written as relu(-d) transposed. 2x2 register
// blocking: 8 fragment loads feed 8 WMMAs per K-step (K=64 in 16 steps).
// ---------------------------------------------------------------------------
__global__ __launch_bounds__(256) void score_kernel(
    const float* __restrict__ M1, const float* __restrict__ M2,
    float* __restrict__ A) {
  const int lane = threadIdx.x & 31;
  const int wave = threadIdx.x >> 5;
  const int wid  = blockIdx.x * 8 + wave;   // 0 .. 256*256-1
  const int bi   = wid >> 8;                // 8192/32 = 256 block rows
  const int bj   = wid & 255;
  if (bi > bj) return;                      // wave-uniform: EXEC stays all-1s

  const int g = lane >> 4;                  // half-wave: K offset +2g
  const int l = lane & 15;                  // row/col within a 16-tile

  const float* rowM1[2]; const float* rowM2[2];
  const float* colM2[2]; const float* colM1[2];
#pragma unroll
  for (int p = 0; p < 2; ++p) {
    rowM1[p] = M1 + (size_t)(bi * 32 + p * 16 + l) * 64;
    rowM2[p] = M2 + (size_t)(bi * 32 + p * 16 + l) * 64;
  }
#pragma unroll
  for (int q = 0; q < 2; ++q) {
    colM2[q] = M2 + (size_t)(bj * 32 + q * 16 + l) * 64;
    colM1[q] = M1 + (size_t)(bj * 32 + q * 16 + l) * 64;
  }

  v8f acc1[2][2] = {{{}, {}}, {{}, {}}};    // S[i][j]  contributions
  v8f acc2[2][2] = {{{}, {}}, {{}, {}}};    // S[j][i]  contributions
#pragma unroll
  for (int k = 0; k < 16; ++k) {
    const int koff = k * 4 + g * 2;         // per-lane K offset (float2)
    v2f a1[2], a2[2], bb1[2], bb2[2];
#pragma unroll
    for (int p = 0; p < 2; ++p) {
      a1[p] = *(const v2f*)(rowM1[p] + koff);
      a2[p] = *(const v2f*)(rowM2[p] + koff);
    }
#pragma unroll
    for (int q = 0; q < 2; ++q) {
      bb1[q] = *(const v2f*)(colM2[q] + koff);
      bb2[q] = *(const v2f*)(colM1[q] + koff);
    }
#pragma unroll
    for (int p = 0; p < 2; ++p)
#pragma unroll
      for (int q = 0; q < 2; ++q) {
        acc1[p][q] = __builtin_amdgcn_wmma_f32_16x16x4_f32(
            false, a1[p], false, bb1[q], (short)0, acc1[p][q], false, false);
        acc2[p][q] = __builtin_amdgcn_wmma_f32_16x16x4_f32(
            false, a2[p], false, bb2[q], (short)0, acc2[p][q], false, false);
      }
  }

#pragma unroll
  for (int p = 0; p < 2; ++p) {
#pragma unroll
    for (int q = 0; q < 2; ++q) {
      float d[8];
#pragma unroll
      for (int r = 0; r < 8; ++r) d[r] = acc1[p][q][r] - acc2[p][q][r];

      // Primary block (bi,bj): lane holds col TJ+l, rows TI+8g+r (coalesced).
      const int TI = bi * 32 + p * 16;
      const int TJ = bj * 32 + q * 16;
      const size_t col = (size_t)TJ + l;
#pragma unroll
      for (int r = 0; r < 8; ++r)
        A[(size_t)(TI + g * 8 + r) * 8192 + col] = fmaxf(d[r], 0.0f);

      // Mirror block (bj,bi) = relu(-d) transposed: 8 consecutive rows of one
      // column -> two aligned float4 stores per lane.
      if (bi != bj) {
        v4f lo, hi;
#pragma unroll
        for (int m = 0; m < 4; ++m) {
          lo[m] = fmaxf(-d[m], 0.0f);
          hi[m] = fmaxf(-d[m + 4], 0.0f);
        }
        float* mbase = A + col * 8192 + (size_t)(TI + g * 8);
        *(v4f*)(mbase)     = lo;
        *(v4f*)(mbase + 4) = hi;
      }
    }
  }
}

// ---------------------------------------------------------------------------
// Kernel 3: per-row top-15 (iterative argmax over LDS-resident row),
// emit edge list and rewrite the masked row in place.
// ---------------------------------------------------------------------------
__global__ __launch_bounds__(256) void topk_kernel(
    float* __restrict__ A, int* __restrict__ ei, float* __restrict__ ea) {
  __shared__ float work[8192];   // 32 KB row copy
  __shared__ float rv[256];
  __shared__ int   ri[256];
  __shared__ float topv[TOPK];
  __shared__ int   topi[TOPK];

  const int t = threadIdx.x;
  const int r = blockIdx.x;
  float* Arow = A + (size_t)r * 8192;

  for (int i = 0; i < 32; ++i) work[t + 256 * i] = Arow[t + 256 * i];
  __syncthreads();

  for (int it = 0; it < TOPK; ++it) {
    float mv = -__builtin_inff();
    int   mi = 0x7fffffff;
    for (int i = 0; i < 32; ++i) {
      const int j = t + 256 * i;              // bank-conflict-free stride
      const float v = work[j];
      if (v > mv || (v == mv && j < mi)) { mv = v; mi = j; }
    }
    rv[t] = mv; ri[t] = mi;
    __syncthreads();
    for (int s = 128; s > 0; s >>= 1) {
      if (t < s) {
        const float ov = rv[t + s]; const int oi = ri[t + s];
        if (ov > rv[t] || (ov == rv[t] && oi < ri[t])) { rv[t] = ov; ri[t] = oi; }
      }
      __syncthreads();
    }
    if (t == 0) {
      topv[it] = rv[0];
      topi[it] = ri[0];
      work[ri[0]] = -__builtin_inff();        // exclude from next extraction
    }
    __syncthreads();
  }

  if (t < TOPK) {
    const int base = r * TOPK + t;
    ei[base]                   = r;           // src row (jnp.repeat(arange(N)))
    ei[N_NODES * TOPK + base]  = topi[t];     // dst = top-k column index
    ea[base]                   = topv[t];     // edge_attr (sorted descending)
  }

  for (int i = 0; i < 32; ++i) {
    const int j = t + 256 * i;
    float v = 0.0f;
#pragma unroll
    for (int m = 0; m < TOPK; ++m)
      if (topi[m] == j) v = topv[m];
    Arow[j] = v;                              // A_masked in place
  }
}

// ---------------------------------------------------------------------------
extern "C" void kernel_launch(void* const* d_in, const int* in_sizes, int n_in,
                              void* d_out, int out_size, void* d_ws, size_t ws_size,
                              hipStream_t stream) {
  const float* x    = (const float*)d_in[0];
  const float* W1   = (const float*)d_in[1];
  const float* b1   = (const float*)d_in[2];
  const float* W2   = (const float*)d_in[3];
  const float* b2   = (const float*)d_in[4];
  const float* prev = (const float*)d_in[5];   // M1

  float* M2 = (float*)d_ws;                    // 8192*64*4 = 2 MB scratch

  // d_out layout (4-byte elems): edge_indices[2][N*TOPK] (int32),
  // edge_attr[N*TOPK] (f32), A_masked[N][N] (f32)
  int*   ei = (int*)d_out;
  float* ea = (float*)d_out + 2 * N_NODES * TOPK;
  float* Am = (float*)d_out + 3 * N_NODES * TOPK;

  mlp_kernel<<<N_NODES / 4, 256, 0, stream>>>(x, W1, b1, W2, b2, M2);
  score_kernel<<<(256 * 256) / 8, 256, 0, stream>>>(prev, M2, Am);
  topk_kernel<<<N_NODES, 256, 0, stream>>>(Am, ei, ea);
}